// FrequencyAwareAttention_3951369912987
// MI455X (gfx1250) — compile-verified
//
#include <hip/hip_runtime.h>

#define DEVI __device__ __forceinline__

typedef float v2f __attribute__((ext_vector_type(2)));
typedef float v8f __attribute__((ext_vector_type(8)));

// Problem dimensions (fixed by the reference)
constexpr int Bb = 16, Tt = 256, Jj = 17, Cc = 512, Hh = 8, HD = 64;
constexpr int Mrows = Bb * Tt * Jj;                  // 69632
constexpr int KEEP  = 64;                            // int((256/2+1)*0.5)
constexpr float SCALE = 0.125f;                      // hd^-0.5
#define TWO_PI_256 0.0245436926061702596f            // 2*pi/256

// ---------------- WMMA helpers (V_WMMA_F32_16X16X4_F32, wave32) -------------
DEVI v8f wmma4(v2f a, v2f b, v8f c) {
  return __builtin_amdgcn_wmma_f32_16x16x4_f32(false, a, false, b, (short)0, c,
                                               false, false);
}
DEVI v8f vzero8() { v8f z = {0.f,0.f,0.f,0.f,0.f,0.f,0.f,0.f}; return z; }

// ------------- CDNA5 async global->LDS copy (ASYNCcnt path) ----------------
// ISA 15.18.3 op 98: GLOBAL_LOAD_ASYNC_TO_LDS_B128  vdst(LDS byte addr), vaddr.
// For LDS apertures the generic pointer's low 32 bits are the LDS byte offset
// (flat addressing).
DEVI void async_ld_b128(void* lds_ptr, const void* gptr) {
  unsigned loff = (unsigned)(size_t)lds_ptr;
  asm volatile("global_load_async_to_lds_b128 %0, %1, off"
               :: "v"(loff), "v"(gptr)
               : "memory");
}
DEVI void wait_async0() {
  asm volatile("s_wait_asynccnt 0x0" ::: "memory");
}

// A-fragment (16x4, M x K), also B-fragment when B is stored N-major ([n][k]).
// Lane L: row = L&15, K pair = k0 + 2*(L>=16).
DEVI v2f ld_a(const float* base, int stride, int k0) {
  int lane = threadIdx.x & 31;
  const float* p = base + (lane & 15) * stride + k0 + ((lane >> 4) << 1);
  v2f f; f.x = p[0]; f.y = p[1]; return f;
}
// B-fragment when B is stored K-major ([k][n]). Lane L: col = n0 + (L&15).
DEVI v2f ld_bk(const float* base, int stride, int k0, int n0) {
  int lane = threadIdx.x & 31;
  int kk = k0 + ((lane >> 4) << 1);
  int n  = n0 + (lane & 15);
  v2f f; f.x = base[kk * stride + n]; f.y = base[(kk + 1) * stride + n];
  return f;
}
// C/D 16x16 f32 tile store: VGPR i holds rows i (lanes 0-15) and i+8 (16-31).
DEVI void st_tile(float* dst, int stride, v8f acc, float mul) {
  int lane = threadIdx.x & 31;
  int n  = lane & 15;
  int mo = (lane >> 4) * 8;
#pragma unroll
  for (int i = 0; i < 8; ++i) dst[(mo + i) * stride + n] = mul * acc[i];
}

// Stage one 64x64 f32 tile (row stride 68 in LDS) via async copies.
// Each of 256 threads moves 4x16B: row = tid>>2, colGroup = (tid&3)*16.
DEVI void stage64x64_async(float* sT, const float* g, int grow_stride,
                           int lr, int lc) {
#pragma unroll
  for (int u = 0; u < 4; ++u) {
    async_ld_b128(&sT[lr * 68 + lc + u * 4],
                  g + (size_t)lr * grow_stride + lc + u * 4);
  }
}

// =================== Kernel 1: QKV projection =============================
// qkv[m, n] = sum_k x[m,k] * w_qkv[n,k];  m=(b*T+t)*J+j, n=(which*8+h)*64+d
// Double-buffered: async-stage chunk i+1 into LDS while WMMAs consume chunk i.
__global__ __launch_bounds__(256) void qkv_gemm(
    const float* __restrict__ x, const float* __restrict__ w,
    float* __restrict__ qb, float* __restrict__ kb, float* __restrict__ vb) {
  __shared__ float As[2][64 * 68];
  __shared__ float Bs[2][64 * 68];
  const int mTile = blockIdx.x * 64, nTile = blockIdx.y * 64;
  const int tid = threadIdx.x, wave = tid >> 5, lane = tid & 31;
  const int mSub  = (wave >> 1) * 16;
  const int nSub0 = (wave & 1) * 32, nSub1 = nSub0 + 16;
  const int lr = tid >> 2, lc = (tid & 3) * 16;
  const float* gx = x + (size_t)mTile * Cc;
  const float* gw = w + (size_t)nTile * Cc;

  v8f acc0 = vzero8(), acc1 = vzero8();
  stage64x64_async(As[0], gx + 0, Cc, lr, lc);
  stage64x64_async(Bs[0], gw + 0, Cc, lr, lc);
  wait_async0();
  __syncthreads();

  int buf = 0;
  for (int kc = 0; kc < Cc; kc += 64) {
    if (kc + 64 < Cc) {
      stage64x64_async(As[buf ^ 1], gx + kc + 64, Cc, lr, lc);
      stage64x64_async(Bs[buf ^ 1], gw + kc + 64, Cc, lr, lc);
    }
#pragma unroll
    for (int k0 = 0; k0 < 64; k0 += 4) {
      v2f a  = ld_a(&As[buf][mSub * 68], 68, k0);
      v2f b0 = ld_a(&Bs[buf][nSub0 * 68], 68, k0);
      v2f b1 = ld_a(&Bs[buf][nSub1 * 68], 68, k0);
      acc0 = wmma4(a, b0, acc0);
      acc1 = wmma4(a, b1, acc1);
    }
    wait_async0();
    __syncthreads();
    buf ^= 1;
  }
  const int mo = (lane >> 4) * 8;
#pragma unroll
  for (int sub = 0; sub < 2; ++sub) {
    v8f acc = sub ? acc1 : acc0;
    int n = nTile + (sub ? nSub1 : nSub0) + (lane & 15);
    int which = n >> 9, c = n & 511, h = c >> 6, d = c & 63;
    float* ob = (which == 0) ? qb : (which == 1) ? kb : vb;
#pragma unroll
    for (int i = 0; i < 8; ++i) {
      int m = mTile + mSub + mo + i;
      int b = m / (Tt * Jj);
      int rem = m - b * (Tt * Jj);
      int t = rem / Jj, j = rem - t * Jj;
      size_t di = ((((size_t)(b * Hh + h) * Jj + j) * Tt + t) * HD + d);
      ob[di] = acc[i];
    }
  }
}

// =================== Kernel 2: time attention =============================
// Per (b,h,j): S = q k^T * scale (256x256), softmax rows, O = S v (256x64).
// All 8 waves cooperate on one 16-query block at a time.
__global__ __launch_bounds__(256) void time_attn(
    const float* __restrict__ q, const float* __restrict__ k,
    const float* __restrict__ v, float* __restrict__ comb) {
  __shared__ float kL[256 * 68];     // keys [key][d]
  __shared__ float pL[16 * 258];     // probs [m][key]
  __shared__ float obuf[4 * 16 * 68];// partial O for K-split combine
  const int bid = blockIdx.x;
  const int b = bid / (Hh * Jj);
  const int hj = bid - b * (Hh * Jj);
  const int h = hj / Jj, j = hj - h * Jj;
  const size_t base = (size_t)bid * Tt * HD;
  const float* qg = q + base;
  const float* kg = k + base;
  const float* vg = v + base;
  const int tid = threadIdx.x, wave = tid >> 5, lane = tid & 31;
  const int mo = (lane >> 4) * 8;

  // prefetch the V panel (consumed from global inside the O-phase WMMAs)
  __builtin_prefetch(vg + (size_t)tid * HD, 0, 3);
  { // async-stage K into LDS (one 64-float row per thread)
#pragma unroll
    for (int u = 0; u < 16; ++u)
      async_ld_b128(&kL[tid * 68 + u * 4], kg + (size_t)tid * HD + u * 4);
  }
  wait_async0();
  __syncthreads();

  for (int qi = 0; qi < 16; ++qi) {
    const int mG = qi * 16;
    // ---- S = q k^T (each wave: 2 key tiles) ----
#pragma unroll
    for (int pp = 0; pp < 2; ++pp) {
      int nt = wave * 2 + pp;
      v8f sacc = vzero8();
#pragma unroll
      for (int k0 = 0; k0 < HD; k0 += 4) {
        v2f a  = ld_a(qg + (size_t)mG * HD, HD, k0);  // 16 q rows from global
        v2f bf = ld_a(&kL[nt * 16 * 68], 68, k0);     // k^T (keys N-major)
        sacc = wmma4(a, bf, sacc);
      }
      st_tile(&pL[nt * 16], 258, sacc, SCALE);
    }
    __syncthreads();
    // ---- softmax over 256 keys; row = tid>>4, 16 elements per thread ----
    {
      int row = tid >> 4, seg = tid & 15;
      float* pr = &pL[row * 258 + seg * 16];
      float vals[16], mx = -1e30f;
#pragma unroll
      for (int u = 0; u < 16; ++u) { vals[u] = pr[u]; mx = fmaxf(mx, vals[u]); }
      for (int off = 1; off < 16; off <<= 1) mx = fmaxf(mx, __shfl_xor(mx, off, 16));
      float s = 0.f;
#pragma unroll
      for (int u = 0; u < 16; ++u) { vals[u] = __expf(vals[u] - mx); s += vals[u]; }
      for (int off = 1; off < 16; off <<= 1) s += __shfl_xor(s, off, 16);
      float inv = 1.f / s;
#pragma unroll
      for (int u = 0; u < 16; ++u) pr[u] = vals[u] * inv;
    }
    __syncthreads();
    // ---- O = P V, K split across wave pairs (kh = wave>>2) ----
    {
      int nt = wave & 3, kh = wave >> 2;
      v8f oacc = vzero8();
      for (int k0 = kh * 128; k0 < kh * 128 + 128; k0 += 4) {
        v2f a  = ld_a(&pL[0], 258, k0);
        // V is K-major in global: v[key][d]
        v2f bf;
        {
          int kk = k0 + ((lane >> 4) << 1);
          int n = nt * 16 + (lane & 15);
          bf.x = vg[(size_t)kk * HD + n];
          bf.y = vg[(size_t)(kk + 1) * HD + n];
        }
        oacc = wmma4(a, bf, oacc);
      }
      if (kh == 1) st_tile(&obuf[nt * 16 * 68], 68, oacc, 1.f);
      __syncthreads();
      if (kh == 0) {
        int d0 = nt * 16 + (lane & 15);
#pragma unroll
        for (int i = 0; i < 8; ++i) {
          float val = oacc[i] + obuf[nt * 16 * 68 + (mo + i) * 68 + (lane & 15)];
          int t = mG + mo + i;
          size_t di = (((size_t)(b * Tt + t) * Jj + j) * Cc + h * HD + d0);
          comb[di] = val;   // raw x_time; gated in freq kernel
        }
      }
    }
  }
}

// =================== Kernel 3: frequency attention ========================
// rfft/irfft realized as GEMMs against on-the-fly twiddle fragments.
__global__ __launch_bounds__(256) void freq_attn(
    const float* __restrict__ q, const float* __restrict__ k,
    const float* __restrict__ v, const float* __restrict__ fgate,
    float* __restrict__ comb) {
  __shared__ float fr[6][64 * 68];   // qr,qi,kr,ki,vr,vi ; fr[0..1] reused as xf
  __shared__ float afr[64 * 68];
  __shared__ float afi[64 * 68];
  const int bid = blockIdx.x;
  const int b = bid / (Hh * Jj);
  const int hj = bid - b * (Hh * Jj);
  const int h = hj / Jj, j = hj - h * Jj;
  const size_t base = (size_t)bid * Tt * HD;
  const float* srcs[3] = {q + base, k + base, v + base};
  const int tid = threadIdx.x, wave = tid >> 5, lane = tid & 31;
  const int mo = (lane >> 4) * 8;

  // prefetch all three 64KB panels (read repeatedly by DFT fragment gathers)
  __builtin_prefetch(srcs[0] + (size_t)tid * HD, 0, 3);
  __builtin_prefetch(srcs[1] + (size_t)tid * HD, 0, 3);
  __builtin_prefetch(srcs[2] + (size_t)tid * HD, 0, 3);

  // ---- Phase A: DFT  f[f,d] = sum_t W[f,t] x[t,d]  (re: cos, im: -sin) ----
  for (int s = 0; s < 3; ++s) {
    const float* src = srcs[s];
#pragma unroll
    for (int pp = 0; pp < 2; ++pp) {
      int p = wave * 2 + pp;
      int mt = p >> 2, nt = p & 3;
      v8f ar = vzero8(), ai = vzero8();
      int f = mt * 16 + (lane & 15);
      for (int k0 = 0; k0 < Tt; k0 += 4) {
        int tt = k0 + ((lane >> 4) << 1);
        v2f ac, as;
        {
          float an0 = TWO_PI_256 * (float)((f * tt) & 255);
          float an1 = TWO_PI_256 * (float)((f * (tt + 1)) & 255);
          ac.x = __cosf(an0); ac.y = __cosf(an1);
          as.x = __sinf(an0); as.y = __sinf(an1);
        }
        v2f bf;
        {
          int n = nt * 16 + (lane & 15);
          bf.x = src[(size_t)tt * HD + n];
          bf.y = src[(size_t)(tt + 1) * HD + n];
        }
        ar = wmma4(ac, bf, ar);
        ai = wmma4(as, bf, ai);
      }
      st_tile(&fr[2 * s][(mt * 16) * 68 + nt * 16], 68, ar, 1.f);
      st_tile(&fr[2 * s + 1][(mt * 16) * 68 + nt * 16], 68, ai, -1.f);
    }
  }
  __syncthreads();

  // ---- Phase B: af = qf * conj(kf)^H over d, scaled ----
#pragma unroll
  for (int pp = 0; pp < 2; ++pp) {
    int p = wave * 2 + pp;
    int mt = p >> 2, nt = p & 3;
    v8f cre = vzero8(), cim = vzero8();
#pragma unroll
    for (int k0 = 0; k0 < KEEP; k0 += 4) {
      v2f aqr = ld_a(&fr[0][(mt * 16) * 68], 68, k0);
      v2f aqi = ld_a(&fr[1][(mt * 16) * 68], 68, k0);
      v2f bkr = ld_a(&fr[2][(nt * 16) * 68], 68, k0);  // kf^T (N-major)
      v2f bki = ld_a(&fr[3][(nt * 16) * 68], 68, k0);
      cre = wmma4(aqr, bkr, cre);
      cre = wmma4(aqi, bki, cre);
      cim = wmma4(aqi, bkr, cim);
      v2f nqr = -aqr;                                  // f32 WMMA has no A-neg
      cim = wmma4(nqr, bki, cim);
    }
    st_tile(&afr[(mt * 16) * 68 + nt * 16], 68, cre, SCALE);
    st_tile(&afi[(mt * 16) * 68 + nt * 16], 68, cim, SCALE);
  }
  __syncthreads();

  // ---- Phase C: softmax over g, real and imag independently ----
  if (tid < 128) {
    float* row = (tid < 64 ? afr : afi) + (tid & 63) * 68;
    float mx = -1e30f;
    for (int u = 0; u < KEEP; ++u) mx = fmaxf(mx, row[u]);
    float s = 0.f;
    for (int u = 0; u < KEEP; ++u) { row[u] = __expf(row[u] - mx); s += row[u]; }
    float inv = 1.f / s;
    for (int u = 0; u < KEEP; ++u) row[u] *= inv;
  }
  __syncthreads();

  // ---- Phase D: xf = af * vf  (complex); write into fr[0] (re), fr[1] (im)
#pragma unroll
  for (int pp = 0; pp < 2; ++pp) {
    int p = wave * 2 + pp;
    int mt = p >> 2, nt = p & 3;
    v8f xr = vzero8(), xi = vzero8();
#pragma unroll
    for (int k0 = 0; k0 < KEEP; k0 += 4) {
      v2f aar = ld_a(&afr[(mt * 16) * 68], 68, k0);
      v2f aai = ld_a(&afi[(mt * 16) * 68], 68, k0);
      v2f bvr = ld_bk(&fr[4][0], 68, k0, nt * 16);     // vf K-major [g][d]
      v2f bvi = ld_bk(&fr[5][0], 68, k0, nt * 16);
      xr = wmma4(aar, bvr, xr);
      v2f nai = -aai;
      xr = wmma4(nai, bvi, xr);
      xi = wmma4(aar, bvi, xi);
      xi = wmma4(aai, bvr, xi);
    }
    st_tile(&fr[0][(mt * 16) * 68 + nt * 16], 68, xr, 1.f);
    st_tile(&fr[1][(mt * 16) * 68 + nt * 16], 68, xi, 1.f);
  }
  __syncthreads();

  // ---- Phase E: irfft (imag of DC ignored, zero-padded freqs drop out),
  //               then gated combine with x_time already in comb ----
  const float g = 1.f / (1.f + __expf(-fgate[0]));
#pragma unroll
  for (int idx = 0; idx < 8; ++idx) {
    int p = wave * 8 + idx;
    int mt = p >> 2, nt = p & 3;
    v8f acc = vzero8();
    int t = mt * 16 + (lane & 15);
    for (int k0 = 0; k0 < KEEP; k0 += 4) {
      int f0 = k0 + ((lane >> 4) << 1);
      v2f acs, asn;
      {
        float an0 = TWO_PI_256 * (float)((t * f0) & 255);
        float an1 = TWO_PI_256 * (float)((t * (f0 + 1)) & 255);
        float c0 = (f0 == 0) ? (1.f / 256.f) : (2.f / 256.f);
        acs.x = c0 * __cosf(an0);
        acs.y = (2.f / 256.f) * __cosf(an1);
        asn.x = (f0 == 0) ? 0.f : (-2.f / 256.f) * __sinf(an0);
        asn.y = (-2.f / 256.f) * __sinf(an1);
      }
      v2f br = ld_bk(&fr[0][0], 68, k0, nt * 16);
      v2f bi = ld_bk(&fr[1][0], 68, k0, nt * 16);
      acc = wmma4(acs, br, acc);
      acc = wmma4(asn, bi, acc);
    }
    int d0 = nt * 16 + (lane & 15);
#pragma unroll
    for (int i = 0; i < 8; ++i) {
      int tr = mt * 16 + mo + i;
      size_t di = (((size_t)(b * Tt + tr) * Jj + j) * Cc + h * HD + d0);
      comb[di] = g * acc[i] + (1.f - g) * comb[di];
    }
  }
}

// =================== Kernel 4: output projection + bias ====================
__global__ __launch_bounds__(256) void out_gemm(
    const float* __restrict__ a, const float* __restrict__ w,
    const float* __restrict__ bp, float* __restrict__ out) {
  __shared__ float As[2][64 * 68];
  __shared__ float Bs[2][64 * 68];
  const int mTile = blockIdx.x * 64, nTile = blockIdx.y * 64;
  const int tid = threadIdx.x, wave = tid >> 5, lane = tid & 31;
  const int mSub  = (wave >> 1) * 16;
  const int nSub0 = (wave & 1) * 32, nSub1 = nSub0 + 16;
  const int lr = tid >> 2, lc = (tid & 3) * 16;
  const float* ga = a + (size_t)mTile * Cc;
  const float* gw = w + (size_t)nTile * Cc;

  v8f acc0 = vzero8(), acc1 = vzero8();
  stage64x64_async(As[0], ga + 0, Cc, lr, lc);
  stage64x64_async(Bs[0], gw + 0, Cc, lr, lc);
  wait_async0();
  __syncthreads();

  int buf = 0;
  for (int kc = 0; kc < Cc; kc += 64) {
    if (kc + 64 < Cc) {
      stage64x64_async(As[buf ^ 1], ga + kc + 64, Cc, lr, lc);
      stage64x64_async(Bs[buf ^ 1], gw + kc + 64, Cc, lr, lc);
    }
#pragma unroll
    for (int k0 = 0; k0 < 64; k0 += 4) {
      v2f af = ld_a(&As[buf][mSub * 68], 68, k0);
      v2f b0 = ld_a(&Bs[buf][nSub0 * 68], 68, k0);
      v2f b1 = ld_a(&Bs[buf][nSub1 * 68], 68, k0);
      acc0 = wmma4(af, b0, acc0);
      acc1 = wmma4(af, b1, acc1);
    }
    wait_async0();
    __syncthreads();
    buf ^= 1;
  }
  const int mo = (lane >> 4) * 8;
#pragma unroll
  for (int sub = 0; sub < 2; ++sub) {
    v8f acc = sub ? acc1 : acc0;
    int n = nTile + (sub ? nSub1 : nSub0) + (lane & 15);
    float bias = bp[n];
#pragma unroll
    for (int i = 0; i < 8; ++i) {
      int m = mTile + mSub + mo + i;
      out[(size_t)m * Cc + n] = acc[i] + bias;
    }
  }
}

// =================== host-side launch =====================================
extern "C" void kernel_launch(void* const* d_in, const int* in_sizes, int n_in,
                              void* d_out, int out_size, void* d_ws,
                              size_t ws_size, hipStream_t stream) {
  (void)in_sizes; (void)n_in; (void)out_size; (void)ws_size;
  const float* x      = (const float*)d_in[0];
  const float* w_qkv  = (const float*)d_in[1];
  const float* w_proj = (const float*)d_in[2];
  const float* b_proj = (const float*)d_in[3];
  const float* fg     = (const float*)d_in[4];
  float* out = (float*)d_out;
  float* ws  = (float*)d_ws;

  const size_t QS = (size_t)Bb * Hh * Jj * Tt * HD;   // 35,651,584 floats
  float* qb   = ws;
  float* kb   = ws + QS;
  float* vb   = ws + 2 * QS;
  float* comb = ws + 3 * QS;                          // (B,T,J,C)

  qkv_gemm<<<dim3(Mrows / 64, (3 * Cc) / 64), 256, 0, stream>>>(
      x, w_qkv, qb, kb, vb);
  time_attn<<<dim3(Bb * Hh * Jj), 256, 0, stream>>>(qb, kb, vb, comb);
  freq_attn<<<dim3(Bb * Hh * Jj), 256, 0, stream>>>(qb, kb, vb, fg, comb);
  out_gemm<<<dim3(Mrows / 64, Cc / 64), 256, 0, stream>>>(
      comb, w_proj, b_proj, out);
}